// VQEmbeddingEMA_74620761801292
// MI455X (gfx1250) — compile-verified
//
#include <hip/hip_runtime.h>
#include <math.h>

// ---------------- problem constants (from reference) ----------------
#define NG   8        // latent groups
#define MCB  512      // codebook size
#define DD   64       // code dim
#define BB   32       // batch
#define LL   1024     // sequence length
#define BL   (BB*LL)  // tokens per group = 32768
#define TOK_PER_BLOCK 256
#define THREADS 256
#define ES_STRIDE 68  // padded LDS row stride (floats): bank-conflict-free WMMA operand loads
#define OUT_ELEMS ((size_t)BB*NG*DD*LL)   // 16777216

typedef __attribute__((ext_vector_type(2))) float v2f;
typedef __attribute__((ext_vector_type(8))) float v8f;

// D = A(16x4) * B(4x16) + C ; fp32 WMMA, chained over K=64 (16 steps)
static __device__ __forceinline__ v8f wmma_f32(v2f a, v2f b, v8f c) {
  return __builtin_amdgcn_wmma_f32_16x16x4_f32(false, a, false, b, (short)0, c,
                                               false, false);
}

__global__ __launch_bounds__(THREADS)
void vq_main_kernel(const float* __restrict__ x,
                    const float* __restrict__ emb,
                    float* __restrict__ out,
                    float* __restrict__ sse_accum,
                    unsigned* __restrict__ counts) {
  // LDS: codebook (512x68) + x tile (256x68) + biases + indices + reduce pad
  __shared__ float es[MCB * ES_STRIDE];          // 139,264 B
  __shared__ float xs[TOK_PER_BLOCK * ES_STRIDE];//  69,632 B
  __shared__ float nhe[MCB];                     //   2,048 B  (-0.5*||e||^2)
  __shared__ int   idxt[TOK_PER_BLOCK];          //   1,024 B
  __shared__ float red[THREADS / 32];

  const int tid = threadIdx.x;
  const int blk = blockIdx.x;            // 0..1023
  const int n   = blk >> 7;              // group
  const int tb  = blk & 127;             // token-block within group
  const int b   = tb >> 2;               // batch index
  const int l0  = (tb & 3) * TOK_PER_BLOCK; // sequence offset

  // ---- stage codebook for this group into LDS (float4, 16B-aligned rows) ----
  {
    const float4* E4 = (const float4*)(emb + (size_t)n * MCB * DD);
    #pragma unroll 4
    for (int i = tid; i < MCB * (DD / 4); i += THREADS) {
      const int m = i >> 4, c = i & 15;
      float4 v = E4[i];
      *(float4*)&es[m * ES_STRIDE + 4 * c] = v;  // 272*m + 16*c : 16B aligned
    }
  }
  // ---- stage x tile: coalesced along l, transposed into [token][d] ----
  {
    const size_t xbase = ((size_t)(b * NG + n) * DD) * LL + (size_t)l0;
    #pragma unroll 4
    for (int d = 0; d < DD; ++d)
      xs[tid * ES_STRIDE + d] = x[xbase + (size_t)d * LL + tid];
  }
  __syncthreads();

  // ---- nhe[m] = -0.5 * ||e_m||^2 ----
  for (int m = tid; m < MCB; m += THREADS) {
    float s = 0.f;
    #pragma unroll
    for (int d = 0; d < DD; ++d) { float v = es[m * ES_STRIDE + d]; s += v * v; }
    nhe[m] = -0.5f * s;
  }
  __syncthreads();

  // ---- per-wave nearest-code search via fp32 WMMA ----
  const int w     = tid >> 5;
  const int lane  = tid & 31;
  const int lm    = lane & 15;
  const int halfk = (lane >> 4) << 1;    // 0 for lanes 0-15, 2 for 16-31

  // A operands: two 16-token sub-tiles, K=64 split into 16 chunks of 4
  v2f a0[16], a1[16];
  {
    const int tok0 = w * 32 + lm;
    const float* r0 = &xs[tok0 * ES_STRIDE + halfk];
    const float* r1 = r0 + 16 * ES_STRIDE;
    #pragma unroll
    for (int kk = 0; kk < 16; ++kk) {
      a0[kk] = *(const v2f*)(r0 + 4 * kk);
      a1[kk] = *(const v2f*)(r1 + 4 * kk);
    }
  }

  float bv0[8], bv1[8]; int bi0[8], bi1[8];
  #pragma unroll
  for (int r = 0; r < 8; ++r) {
    bv0[r] = -INFINITY; bv1[r] = -INFINITY; bi0[r] = 0; bi1[r] = 0;
  }

  for (int c = 0; c < MCB / 16; ++c) {   // 32 codebook chunks of 16 codes
    const int code = c * 16 + lm;
    const float bias = nhe[code];        // same code for all 8 C-rows in lane
    v8f acc0 = {bias, bias, bias, bias, bias, bias, bias, bias};
    v8f acc1 = acc0;
    const float* eb = &es[code * ES_STRIDE + halfk];
    #pragma unroll
    for (int kk = 0; kk < 16; ++kk) {
      v2f bm = *(const v2f*)(eb + 4 * kk);    // conflict-free b64 LDS load
      acc0 = wmma_f32(a0[kk], bm, acc0);
      acc1 = wmma_f32(a1[kk], bm, acc1);
    }
    #pragma unroll
    for (int r = 0; r < 8; ++r) {
      if (acc0[r] > bv0[r]) { bv0[r] = acc0[r]; bi0[r] = code; }
      if (acc1[r] > bv1[r]) { bv1[r] = acc1[r]; bi1[r] = code; }
    }
  }

  // ---- butterfly argmax across each 16-lane half (tie -> smaller index) ----
  #pragma unroll
  for (int r = 0; r < 8; ++r) {
    float v = bv0[r]; int i = bi0[r];
    #pragma unroll
    for (int m = 1; m < 16; m <<= 1) {
      float ov = __shfl_xor(v, m, 32); int oi = __shfl_xor(i, m, 32);
      if (ov > v || (ov == v && oi < i)) { v = ov; i = oi; }
    }
    if (lane == 0)  idxt[w * 32 + r]     = i;   // token rows 0..7
    if (lane == 16) idxt[w * 32 + 8 + r] = i;   // token rows 8..15

    v = bv1[r]; i = bi1[r];
    #pragma unroll
    for (int m = 1; m < 16; m <<= 1) {
      float ov = __shfl_xor(v, m, 32); int oi = __shfl_xor(i, m, 32);
      if (ov > v || (ov == v && oi < i)) { v = ov; i = oi; }
    }
    if (lane == 0)  idxt[w * 32 + 16 + r] = i;
    if (lane == 16) idxt[w * 32 + 24 + r] = i;
  }
  __syncthreads();

  // ---- fused epilogue: gather quantized rows, write out, sse, histogram ----
  const int myidx = idxt[tid];
  const float* er = &es[myidx * ES_STRIDE];
  const float* xr = &xs[tid * ES_STRIDE];
  const size_t obase = ((size_t)(b * NG + n) * DD) * LL + (size_t)l0 + tid;
  float sse = 0.f;
  #pragma unroll 4
  for (int d = 0; d < DD; ++d) {
    const float q  = er[d];
    const float df = xr[d] - q;
    sse += df * df;
    out[obase + (size_t)d * LL] = q;     // coalesced along l
  }
  atomicAdd(&counts[n * MCB + myidx], 1u);

  #pragma unroll
  for (int m = 16; m >= 1; m >>= 1) sse += __shfl_xor(sse, m, 32);
  if (lane == 0) red[w] = sse;
  __syncthreads();
  if (tid == 0) {
    float s = 0.f;
    #pragma unroll
    for (int i = 0; i < THREADS / 32; ++i) s += red[i];
    atomicAdd(sse_accum, s);
  }
}

__global__ __launch_bounds__(THREADS)
void vq_finalize_kernel(const float* __restrict__ sse_accum,
                        const unsigned* __restrict__ counts,
                        float* __restrict__ scalars) {
  __shared__ float gs[NG];
  const int tid = threadIdx.x;
  const int w = tid >> 5, lane = tid & 31;   // wave w handles group n=w
  float s = 0.f;
  #pragma unroll
  for (int j = 0; j < MCB / 32; ++j) {
    const float p = (float)counts[w * MCB + lane + 32 * j] * (1.0f / (float)BL);
    s += p * logf(p + 1e-10f);
  }
  #pragma unroll
  for (int m = 16; m >= 1; m >>= 1) s += __shfl_xor(s, m, 32);
  if (lane == 0) gs[w] = expf(-s);
  __syncthreads();
  if (tid == 0) {
    float perp = 0.f;
    #pragma unroll
    for (int i = 0; i < NG; ++i) perp += gs[i];
    scalars[0] = 0.25f * (*sse_accum) / (float)OUT_ELEMS;  // commitment loss
    scalars[1] = perp;                                     // perplexity
  }
}

extern "C" void kernel_launch(void* const* d_in, const int* in_sizes, int n_in,
                              void* d_out, int out_size, void* d_ws, size_t ws_size,
                              hipStream_t stream) {
  const float* x   = (const float*)d_in[0];   // [B, N*D, L] fp32
  const float* emb = (const float*)d_in[1];   // [N, M, D]   fp32
  float* out = (float*)d_out;                 // [B*N*D*L] + loss + perplexity

  float*    sse    = (float*)d_ws;
  unsigned* counts = (unsigned*)((char*)d_ws + 64);

  hipMemsetAsync(d_ws, 0, 64 + (size_t)NG * MCB * sizeof(unsigned), stream);

  vq_main_kernel<<<dim3(NG * (BL / TOK_PER_BLOCK)), dim3(THREADS), 0, stream>>>(
      x, emb, out, sse, counts);
  vq_finalize_kernel<<<dim3(1), dim3(THREADS), 0, stream>>>(
      sse, counts, out + OUT_ELEMS);
}